// GraphSAGE_34196529611151
// MI455X (gfx1250) — compile-verified
//
#include <hip/hip_runtime.h>

// GraphSAGE on MI455X (gfx1250, wave32).
// Bottleneck analysis: edge gather/scatter (~0.8 GB L2 traffic per layer, all
// L2-resident on 192MB L2). GEMMs (~4.9 GFLOP total) are negligible -> use
// exact-precision f32 WMMA (v_wmma_f32_16x16x4_f32) for the dense math.

typedef __attribute__((ext_vector_type(2))) float v2f;
typedef __attribute__((ext_vector_type(8))) float v8f;

#define N_NODES 50000
#define N_EDGES 800000
#define IN_FEATS 128
#define HID_FEATS 128
#define OUT_FEATS 64

// ---------------------------------------------------------------------------
// Zero a float range (grid-stride).
__global__ void zero_f32(float* __restrict__ p, size_t n) {
    size_t i = (size_t)blockIdx.x * blockDim.x + threadIdx.x;
    size_t stride = (size_t)gridDim.x * blockDim.x;
    for (; i < n; i += stride) p[i] = 0.0f;
}

// ---------------------------------------------------------------------------
// deg[dst[e]] += 1.0f
__global__ void degree_kernel(const int* __restrict__ dst, float* __restrict__ deg, int E) {
    int i = blockIdx.x * blockDim.x + threadIdx.x;
    if (i < E) atomicAdd(&deg[dst[i]], 1.0f);
}

// deg -> 1/max(deg,1)  (isolated nodes => mean contribution 0 since agg==0)
__global__ void invdeg_kernel(float* __restrict__ deg, int n) {
    int i = blockIdx.x * blockDim.x + threadIdx.x;
    if (i < n) deg[i] = 1.0f / fmaxf(deg[i], 1.0f);
}

// ---------------------------------------------------------------------------
// Edge scatter: one wave per edge. Lane l moves features [4l, 4l+3] of
// feat[src[e]] into agg[dst[e]] with global f32 atomics (512B per edge).
__global__ void scatter_kernel(const float* __restrict__ feat,
                               const int* __restrict__ src,
                               const int* __restrict__ dst,
                               float* __restrict__ agg, int E) {
    const int lane = threadIdx.x & 31;
    const int e = (int)(((size_t)blockIdx.x * blockDim.x + threadIdx.x) >> 5);
    if (e >= E) return;
    const int s = src[e];   // wave-uniform -> scalarized
    const int d = dst[e];
    const float4 v = ((const float4*)(feat + (size_t)s * 128))[lane];
    float* ap = agg + (size_t)d * 128 + lane * 4;
    atomicAdd(ap + 0, v.x);
    atomicAdd(ap + 1, v.y);
    atomicAdd(ap + 2, v.z);
    atomicAdd(ap + 3, v.w);
}

// ---------------------------------------------------------------------------
// Fused SAGE layer GEMM:
//   Y = act( X @ Wself + (AGG * invdeg) @ Wneigh + bias )
// One wave computes one 16x16 output tile via v_wmma_f32_16x16x4_f32.
// K = 128 (compile-time), NF = output feature count (128 or 64).
//
// Operand layouts per CDNA5 ISA (wave32):
//   A (16x4 f32, 2 VGPRs): lanes 0-15 -> M=lane,   v0=K0, v1=K1
//                          lanes 16-31 -> M=lane-16, v0=K2, v1=K3
//   B (4x16 f32, 2 VGPRs): mirrored with N=lane&15
//   C/D (16x16 f32, 8 VGPRs): reg i, lanes 0-15: (M=i,  N=lane)
//                                      lanes 16-31: (M=i+8, N=lane-16)
template <int NF, bool RELU>
__global__ void sage_gemm(const float* __restrict__ X,
                          const float* __restrict__ AGG,
                          const float* __restrict__ invdeg,
                          const float* __restrict__ Wself,
                          const float* __restrict__ Wneigh,
                          const float* __restrict__ bias,
                          float* __restrict__ Y) {
    constexpr int K = 128;
    const int lane = threadIdx.x & 31;
    const int wv   = threadIdx.x >> 5;
    const int mBase = blockIdx.x * 16;
    const int nBase = wv * 16;
    if (nBase >= NF) return;

    const int half = lane >> 4;     // 0 or 1
    const int l16  = lane & 15;
    const int aRow = mBase + l16;   // A-matrix row handled by this lane
    const int kOff = half * 2;      // lanes 16-31 hold K+2 / K+3
    const int bCol = nBase + l16;   // B/C/D column handled by this lane

    const float idg = invdeg[aRow];
    const float* __restrict__ xrow = X   + (size_t)aRow * K;
    const float* __restrict__ grow = AGG + (size_t)aRow * K;

    v8f acc = {};
#pragma unroll
    for (int kk = 0; kk < K; kk += 4) {
        const int k0 = kk + kOff;
        const float2 xs = *(const float2*)(xrow + k0);   // 8B aligned (k0 even)
        const float2 gs = *(const float2*)(grow + k0);
        v2f a_s, a_n, b_s, b_n;
        a_s[0] = xs.x;        a_s[1] = xs.y;
        a_n[0] = gs.x * idg;  a_n[1] = gs.y * idg;       // fold mean into load
        b_s[0] = Wself[(size_t)k0 * NF + bCol];
        b_s[1] = Wself[(size_t)(k0 + 1) * NF + bCol];
        b_n[0] = Wneigh[(size_t)k0 * NF + bCol];
        b_n[1] = Wneigh[(size_t)(k0 + 1) * NF + bCol];
        acc = __builtin_amdgcn_wmma_f32_16x16x4_f32(false, a_s, false, b_s,
                                                    (short)0, acc, false, false);
        acc = __builtin_amdgcn_wmma_f32_16x16x4_f32(false, a_n, false, b_n,
                                                    (short)0, acc, false, false);
    }

    const float bb = bias[bCol];
#pragma unroll
    for (int i = 0; i < 8; ++i) {
        float v = acc[i] + bb;
        if (RELU) v = fmaxf(v, 0.0f);
        const int row = mBase + i + half * 8;
        Y[(size_t)row * NF + bCol] = v;
    }
}

// ---------------------------------------------------------------------------
extern "C" void kernel_launch(void* const* d_in, const int* in_sizes, int n_in,
                              void* d_out, int out_size, void* d_ws, size_t ws_size,
                              hipStream_t stream) {
    (void)in_sizes; (void)n_in; (void)out_size; (void)ws_size;

    const float* x   = (const float*)d_in[0];
    const int*   src = (const int*)d_in[1];
    const int*   dst = (const int*)d_in[2];
    const float* W1s = (const float*)d_in[3];
    const float* W1n = (const float*)d_in[4];
    const float* b1  = (const float*)d_in[5];
    const float* W2s = (const float*)d_in[6];
    const float* W2n = (const float*)d_in[7];
    const float* b2  = (const float*)d_in[8];
    float* out = (float*)d_out;

    // Workspace layout (all f32):
    //   agg    : N_NODES*128  (reused by both layers)
    //   h      : N_NODES*128  (layer-1 activations)
    //   invdeg : N_NODES
    float* agg    = (float*)d_ws;
    float* h      = agg + (size_t)N_NODES * 128;
    float* invdeg = h   + (size_t)N_NODES * 128;

    const size_t aggElems = (size_t)N_NODES * 128;
    const int ZB = 4096;                       // zero-kernel blocks
    const int scatterBlocks = (N_EDGES * 32 + 255) / 256;   // one wave per edge
    const int mTiles = N_NODES / 16;           // 3125 exactly

    // ---- degree / mean prep -------------------------------------------------
    zero_f32<<<ZB, 256, 0, stream>>>(agg, aggElems);
    zero_f32<<<(N_NODES + 255) / 256, 256, 0, stream>>>(invdeg, (size_t)N_NODES);
    degree_kernel<<<(N_EDGES + 255) / 256, 256, 0, stream>>>(dst, invdeg, N_EDGES);
    invdeg_kernel<<<(N_NODES + 255) / 256, 256, 0, stream>>>(invdeg, N_NODES);

    // ---- layer 1: h = relu(x@W1s + mean@W1n + b1) ---------------------------
    scatter_kernel<<<scatterBlocks, 256, 0, stream>>>(x, src, dst, agg, N_EDGES);
    sage_gemm<HID_FEATS, true><<<mTiles, 256, 0, stream>>>(x, agg, invdeg,
                                                           W1s, W1n, b1, h);

    // ---- layer 2: out = h@W2s + mean(h)@W2n + b2 ----------------------------
    zero_f32<<<ZB, 256, 0, stream>>>(agg, aggElems);
    scatter_kernel<<<scatterBlocks, 256, 0, stream>>>(h, src, dst, agg, N_EDGES);
    sage_gemm<OUT_FEATS, false><<<mTiles, 128, 0, stream>>>(h, agg, invdeg,
                                                            W2s, W2n, b2, out);
}